// GraphFlow_GCN_22471268892731
// MI455X (gfx1250) — compile-verified
//
#include <hip/hip_runtime.h>
#include <cstdint>
#include <cstddef>

#define NN 100000
#define NE 1600000

typedef __attribute__((ext_vector_type(2))) float v2f;
typedef __attribute__((ext_vector_type(8))) float v8f;

// ---------- preprocessing kernels ----------

__global__ void deg_kernel(const long long* __restrict__ col,
                           const float* __restrict__ ew,
                           float* __restrict__ deg) {
    int e = blockIdx.x * blockDim.x + threadIdx.x;
    if (e >= NE) return;
    atomicAdd(&deg[(int)col[e]], ew[e]);
}

__global__ void dis_kernel(float* __restrict__ deg) {
    int n = blockIdx.x * blockDim.x + threadIdx.x;
    if (n >= NN) return;
    deg[n] = rsqrtf(deg[n] + 1.0f);   // deg>=1 always (edge_attr >= 0)
}

__global__ void norm_kernel(const long long* __restrict__ row,
                            const long long* __restrict__ col,
                            const float* __restrict__ ew,
                            const float* __restrict__ dis,
                            float* __restrict__ nrm) {
    int e = blockIdx.x * blockDim.x + threadIdx.x;
    if (e >= NE) return;
    nrm[e] = dis[(int)row[e]] * ew[e] * dis[(int)col[e]];
}

__global__ void build_h0(const float* __restrict__ t,
                         const float* __restrict__ data,
                         float* __restrict__ h0) {
    int tid = blockIdx.x * blockDim.x + threadIdx.x;
    if (tid >= NN * 64) return;
    int n = tid >> 6;
    int f = tid & 63;
    h0[tid] = (f == 0) ? t[0] : data[n * 63 + (f - 1)];
}

// ---------- WMMA f32 GEMM: X[16-node tile, 16-col tile] = H @ W ----------
// One wave (32 lanes) per 16x16 output tile; K stepped by 4 via
// V_WMMA_F32_16X16X4_F32. Exact fp32 — matches reference numerics.
// A frag (16x4): lane l -> row m=l&15, K pair kb=(l>>4)*2 in v[0],v[1].
// B frag (4x16): lane l -> col m=l&15, same K pair (mirrored layout).
// D (16x16 f32, 8 VGPRs): lane l -> col l&15, rows (l>>4)*8 + i.
__global__ void __launch_bounds__(32)
gemm_wmma_f32(const float* __restrict__ H, const float* __restrict__ W,
              float* __restrict__ X, int K, int ldH, int outDim, int ldX) {
    const int l   = threadIdx.x;
    const int m   = l & 15;
    const int kb  = (l >> 4) << 1;
    const int nodeBase = blockIdx.x << 4;
    const int co  = (blockIdx.y << 4) + m;
    // branchless pad handling (keeps EXEC all-ones around WMMA)
    const int   cosafe = (co < outDim) ? co : (outDim - 1);
    const float cmask  = (co < outDim) ? 1.0f : 0.0f;

    const float* arow = H + (size_t)(nodeBase + m) * ldH;
    v8f acc = {};
    for (int k = 0; k < K; k += 4) {
        v2f a, b;
        a.x = arow[k + kb];
        a.y = arow[k + kb + 1];
        b.x = W[(size_t)(k + kb)     * outDim + cosafe] * cmask;
        b.y = W[(size_t)(k + kb + 1) * outDim + cosafe] * cmask;
        acc = __builtin_amdgcn_wmma_f32_16x16x4_f32(
            /*neg_a=*/false, a, /*neg_b=*/false, b,
            /*c_mod=*/(short)0, acc, /*reuse_a=*/false, /*reuse_b=*/false);
    }
    const int mrowBase = (l >> 4) << 3;
#pragma unroll
    for (int i = 0; i < 8; ++i)
        X[(size_t)(nodeBase + mrowBase + i) * ldX + co] = acc[i];
}

// ---------- edge scatter: AGG[col] += X[row] * norm ----------
// one thread per (edge, 4-feature chunk); chunk count is a power of two.
__global__ void scatter_kernel(const long long* __restrict__ row,
                               const long long* __restrict__ col,
                               const float* __restrict__ nrm,
                               const float* __restrict__ X,
                               float* __restrict__ AGG,
                               int f4shift, int ld) {
    long long tid = (long long)blockIdx.x * blockDim.x + threadIdx.x;
    long long total = (long long)NE << f4shift;
    if (tid >= total) return;
    int e = (int)(tid >> f4shift);
    int c = ((int)tid & ((1 << f4shift) - 1)) << 2;
    int r  = (int)row[e];
    int cn = (int)col[e];
    float nv = nrm[e];
    const float4 xv = *(const float4*)(X + (size_t)r * ld + c);
    float* dst = AGG + (size_t)cn * ld + c;
    atomicAdd(dst + 0, xv.x * nv);
    atomicAdd(dst + 1, xv.y * nv);
    atomicAdd(dst + 2, xv.z * nv);
    atomicAdd(dst + 3, xv.w * nv);
}

// ---------- finalize: DST = AGG + X*dis^2 + bias, optional tanh ----------
// grid.x = node, threadIdx.x = feature (<= 64). May write DST==AGG in place.
__global__ void finalize_kernel(const float* AGG, const float* __restrict__ X,
                                const float* __restrict__ dis,
                                const float* __restrict__ bias,
                                float* DST, int ldIn, int outF, int ldOut,
                                int doTanh) {
    int n = blockIdx.x;
    int f = threadIdx.x;
    if (f >= outF) return;
    float sn = dis[n]; sn *= sn;
    float v = AGG[(size_t)n * ldIn + f] + X[(size_t)n * ldIn + f] * sn + bias[f];
    if (doTanh) v = tanhf(v);
    DST[(size_t)n * ldOut + f] = v;
}

// ---------- host launcher ----------
extern "C" void kernel_launch(void* const* d_in, const int* in_sizes, int n_in,
                              void* d_out, int out_size, void* d_ws, size_t ws_size,
                              hipStream_t stream) {
    const float*     t     = (const float*)d_in[0];
    const float*     data  = (const float*)d_in[1];
    const long long* edges = (const long long*)d_in[2];   // int64 [2, NE]
    // d_in[3] = pos (unused by reference path)
    const float*     ew    = (const float*)d_in[4];
    const float*     W1    = (const float*)d_in[5];
    const float*     b1    = (const float*)d_in[6];
    const float*     W2    = (const float*)d_in[7];
    const float*     b2    = (const float*)d_in[8];
    const float*     W3    = (const float*)d_in[9];
    const float*     b3    = (const float*)d_in[10];

    const long long* row = edges;        // source j
    const long long* col = edges + NE;   // target i

    float* ws   = (float*)d_ws;
    float* dis  = ws;                              // NN floats (deg -> dis in place)
    float* nrm  = dis + NN;                        // NE floats
    float* bufA = nrm + NE;                        // NN*64 floats
    float* bufB = bufA + (size_t)NN * 64;          // NN*64 floats

    // --- degree / norm precompute ---
    hipMemsetAsync(dis, 0, (size_t)NN * sizeof(float), stream);
    deg_kernel <<<(NE + 255) / 256, 256, 0, stream>>>(col, ew, dis);
    dis_kernel <<<(NN + 255) / 256, 256, 0, stream>>>(dis);
    norm_kernel<<<(NE + 255) / 256, 256, 0, stream>>>(row, col, ew, dis, nrm);

    // --- h0 = [t | data], [NN, 64] ---
    build_h0<<<(NN * 64 + 255) / 256, 256, 0, stream>>>(t, data, bufA);

    // --- layer 1: 64 -> 64, tanh ---
    gemm_wmma_f32<<<dim3(NN / 16, 4), 32, 0, stream>>>(bufA, W1, bufB, 64, 64, 64, 64);
    hipMemsetAsync(bufA, 0, (size_t)NN * 64 * sizeof(float), stream);
    {
        long long tot = (long long)NE << 4;
        scatter_kernel<<<(unsigned)((tot + 255) / 256), 256, 0, stream>>>(
            row, col, nrm, bufB, bufA, 4, 64);
    }
    finalize_kernel<<<NN, 64, 0, stream>>>(bufA, bufB, dis, b1, bufA, 64, 64, 64, 1);

    // --- layer 2: 64 -> 32, tanh ---
    gemm_wmma_f32<<<dim3(NN / 16, 2), 32, 0, stream>>>(bufA, W2, bufB, 64, 64, 32, 32);
    hipMemsetAsync(bufA, 0, (size_t)NN * 32 * sizeof(float), stream);
    {
        long long tot = (long long)NE << 3;
        scatter_kernel<<<(unsigned)((tot + 255) / 256), 256, 0, stream>>>(
            row, col, nrm, bufB, bufA, 3, 32);
    }
    finalize_kernel<<<NN, 32, 0, stream>>>(bufA, bufB, dis, b2, bufA, 32, 32, 32, 1);

    // --- layer 3: 32 -> 63 (padded to stride 64), no tanh, write d_out ---
    gemm_wmma_f32<<<dim3(NN / 16, 4), 32, 0, stream>>>(bufA, W3, bufB, 32, 32, 63, 64);
    hipMemsetAsync(bufA, 0, (size_t)NN * 64 * sizeof(float), stream);
    {
        long long tot = (long long)NE << 4;   // pad col 63 holds zeros; adds are no-ops
        scatter_kernel<<<(unsigned)((tot + 255) / 256), 256, 0, stream>>>(
            row, col, nrm, bufB, bufA, 4, 64);
    }
    finalize_kernel<<<NN, 64, 0, stream>>>(bufA, bufB, dis, b3, (float*)d_out,
                                           64, 63, 63, 0);
}